// LearnableLogOptimalTransport_38465727103432
// MI455X (gfx1250) — compile-verified
//
#include <hip/hip_runtime.h>
#include <stdint.h>

// LearnableLogOptimalTransport (log-domain Sinkhorn) for MI455X (gfx1250).
// B=32, M=N=1024, 100 iterations. Z (134.5MB) is never materialized: it is
// recomputed from scores + alpha, so the working set stays resident in the
// 192MB L2 and both passes per iteration run at L2 bandwidth.
//
// Col pass: async global->LDS tile staging (ASYNCcnt) + v_wmma_f32_16x16x4_f32
//           column-sum accumulation (matrix pipe co-executes with v_exp_f32).
// Row pass: wave-per-row coalesced b128 streaming + wave32 shfl reduction.

#define B_ 32
#define M_ 1024
#define N_ 1024
#define INF_ 1e12f
#define LOG2E_ 1.4426950408889634f
#define LN2_ 0.6931471805599453f

typedef float v2f __attribute__((ext_vector_type(2)));
typedef float v8f __attribute__((ext_vector_type(8)));
typedef int   v4i __attribute__((ext_vector_type(4)));
typedef v4i __attribute__((address_space(1)))* gv4ip;   // global int4*
typedef v4i __attribute__((address_space(3)))* lv4ip;   // LDS int4*

#if defined(__has_builtin)
#if __has_builtin(__builtin_amdgcn_wmma_f32_16x16x4_f32)
#define HAVE_WMMA4 1
#endif
#if __has_builtin(__builtin_amdgcn_global_load_async_to_lds_b128) && \
    __has_builtin(__builtin_amdgcn_s_wait_asynccnt)
#define HAVE_ASYNC 1
#endif
#endif

__device__ __forceinline__ float fexp(float x) {            // exp(x)
    return __builtin_amdgcn_exp2f(x * LOG2E_);              // v_exp_f32
}
__device__ __forceinline__ float flog(float x) {            // ln(x)
    return __builtin_amdgcn_logf(x) * LN2_;                 // v_log_f32
}

// ---------------------------------------------------------------- init ----
__global__ __launch_bounds__(256)
void ot_init(const unsigned char* __restrict__ rowm,
             const unsigned char* __restrict__ colm,
             float* __restrict__ u, float* __restrict__ v,
             float* __restrict__ logmu, float* __restrict__ lognu,
             float* __restrict__ normv) {
    int b = blockIdx.x, tid = threadIdx.x;
    __shared__ float red[256];
    float cr = 0.f, cc = 0.f;
    for (int i = tid; i < M_; i += 256) cr += rowm[b * M_ + i] ? 1.f : 0.f;
    for (int i = tid; i < N_; i += 256) cc += colm[b * N_ + i] ? 1.f : 0.f;
    red[tid] = cr; __syncthreads();
    for (int s = 128; s > 0; s >>= 1) { if (tid < s) red[tid] += red[tid + s]; __syncthreads(); }
    float nr = red[0]; __syncthreads();
    red[tid] = cc; __syncthreads();
    for (int s = 128; s > 0; s >>= 1) { if (tid < s) red[tid] += red[tid + s]; __syncthreads(); }
    float nc = red[0];
    float nrm = -flog(nr + nc);
    if (tid == 0) normv[b] = nrm;
    for (int i = tid; i <= M_; i += 256) {
        float lm = (i < M_) ? (rowm[b * M_ + i] ? nrm : -INF_) : (flog(nc) + nrm);
        logmu[(size_t)b * (M_ + 1) + i] = lm;
        u[(size_t)b * (M_ + 1) + i] = 0.f;
    }
    for (int i = tid; i <= N_; i += 256) {
        float ln = (i < N_) ? (colm[b * N_ + i] ? nrm : -INF_) : (flog(nr) + nrm);
        lognu[(size_t)b * (N_ + 1) + i] = ln;
        v[(size_t)b * (N_ + 1) + i] = 0.f;
    }
}

// ------------------------------------------------------------ row pass ----
// u[b,m] = log_mu[b,m] - log( sum_n exp(Z[b,m,n] + v[b,n]) )
// One wave per row; col masks folded into staged v (invalid -> v-1e12 -> exp 0).
__global__ __launch_bounds__(256)
void ot_row_pass(const float* __restrict__ scores,
                 const unsigned char* __restrict__ rowm,
                 const unsigned char* __restrict__ colm,
                 const float* __restrict__ alpha_p,
                 const float* __restrict__ v,
                 const float* __restrict__ logmu,
                 float* __restrict__ u) {
    int b = blockIdx.y;
    __shared__ __align__(16) float vl[N_ + 8];
    const float* vb = v + (size_t)b * (N_ + 1);
    for (int i = threadIdx.x; i <= N_; i += 256) {
        float vv = vb[i];
        if (i < N_ && !colm[b * N_ + i]) vv -= INF_;
        vl[i] = vv;
    }
    __syncthreads();
    float alpha = *alpha_p;
    int wave = threadIdx.x >> 5, lane = threadIdx.x & 31;
    if (blockIdx.x < M_ / 8) {
        int m = blockIdx.x * 8 + wave;
        const float* srow = scores + ((size_t)b * M_ + m) * N_;
        float rb = rowm[b * M_ + m] ? 0.f : -INF_;
        float acc = 0.f;
#pragma unroll
        for (int i = 0; i < 8; ++i) {              // 8 x coalesced b128 per wave
            int n = i * 128 + lane * 4;
            float4 s = *(const float4*)(srow + n);
            float4 w = *(const float4*)(&vl[n]);
            acc += fexp(s.x + rb + w.x);
            acc += fexp(s.y + rb + w.y);
            acc += fexp(s.z + rb + w.z);
            acc += fexp(s.w + rb + w.w);
        }
        for (int o = 16; o > 0; o >>= 1) acc += __shfl_xor(acc, o, 32);
        if (lane == 0) {
            acc += fexp(alpha + rb + vl[N_]);      // dustbin column
            u[(size_t)b * (M_ + 1) + m] =
                logmu[(size_t)b * (M_ + 1) + m] - flog(acc);
        }
    } else if (wave == 0) {                         // dustbin row m = M
        float acc = 0.f;
        for (int n = lane; n <= N_; n += 32) acc += fexp(alpha + vl[n]);
        for (int o = 16; o > 0; o >>= 1) acc += __shfl_xor(acc, o, 32);
        if (lane == 0)
            u[(size_t)b * (M_ + 1) + M_] =
                logmu[(size_t)b * (M_ + 1) + M_] - flog(acc);
    }
}

// ------------------------------------------------------------ col pass ----
// v[b,n] = log_nu[b,n] - log( sum_m exp(Z[b,m,n] + u[b,m]) )
// 32x128 score tiles double-buffered via async global->LDS; per-wave 16
// columns accumulated on the matrix pipe: D = ones(16x4) x E(4x16) + C gives
// column sums replicated in every D row.
#define TROWS 32
#define TS    136   // padded LDS row stride (floats): half-waves hit disjoint banks

__global__ __launch_bounds__(256)
void ot_col_pass(const float* __restrict__ scores,
                 const unsigned char* __restrict__ rowm,
                 const unsigned char* __restrict__ colm,
                 const float* __restrict__ alpha_p,
                 const float* __restrict__ u,
                 const float* __restrict__ lognu,
                 float* __restrict__ v) {
    int b = blockIdx.y;
    __shared__ __align__(16) float ul[M_ + 8];
    __shared__ __align__(16) float tile[2 * TROWS * TS];   // 34816 B
    const float* ub = u + (size_t)b * (M_ + 1);
    for (int i = threadIdx.x; i <= M_; i += 256) {
        float uu = ub[i];
        if (i < M_ && !rowm[b * M_ + i]) uu -= INF_;
        ul[i] = uu;
    }
    __syncthreads();
    float alpha = *alpha_p;
    int wave = threadIdx.x >> 5, lane = threadIdx.x & 31;

    if (blockIdx.x < N_ / 128) {
        int colBase = blockIdx.x * 128;
        const float* sb = scores + (size_t)b * M_ * N_ + colBase;

        auto prefetch = [&](int chunk, int buf) {
            int m0 = chunk * TROWS;
#pragma unroll
            for (int j = 0; j < 4; ++j) {           // 4 async b128 per thread
                int f = (int)threadIdx.x + 256 * j;
                int row = f >> 5, cq = f & 31;      // 32 float4 per tile row
                const float* g = sb + (size_t)(m0 + row) * N_ + cq * 4;
                float* l = &tile[buf * TROWS * TS + row * TS + cq * 4];
#ifdef HAVE_ASYNC
                __builtin_amdgcn_global_load_async_to_lds_b128(
                    (gv4ip)g, (lv4ip)l, 0, 0);
#else
                *(float4*)l = *(const float4*)g;
#endif
            }
        };

        const int NCHUNK = M_ / TROWS;              // 32
        int kk = (lane >> 4) * 2;                   // K rows owned by this half-wave
        int ncol = wave * 16 + (lane & 15);         // column within 128-tile
#ifdef HAVE_WMMA4
        v8f c = {};
        const v2f ones = {1.f, 1.f};
#else
        float accA = 0.f;
#endif
        prefetch(0, 0);
        for (int chunk = 0; chunk < NCHUNK; ++chunk) {
            if (chunk + 1 < NCHUNK) {
                prefetch(chunk + 1, (chunk + 1) & 1);
#ifdef HAVE_ASYNC
                __builtin_amdgcn_s_wait_asynccnt(4);   // chunk's 4 loads done
#endif
            } else {
#ifdef HAVE_ASYNC
                __builtin_amdgcn_s_wait_asynccnt(0);
#endif
            }
            __syncthreads();                        // tile for `chunk` visible
            const float* tb = &tile[(chunk & 1) * TROWS * TS];
            int mg = chunk * TROWS;
#pragma unroll
            for (int sub = 0; sub < 8; ++sub) {     // 4 rows (K=4) per step
                int r = sub * 4 + kk;
                float e0 = fexp(tb[r * TS + ncol] + ul[mg + r]);
                float e1 = fexp(tb[(r + 1) * TS + ncol] + ul[mg + r + 1]);
#ifdef HAVE_WMMA4
                v2f bm = {e0, e1};
                c = __builtin_amdgcn_wmma_f32_16x16x4_f32(
                        false, ones, false, bm, (short)0, c, false, false);
#else
                accA += e0 + e1;
#endif
            }
            __syncthreads();                        // done reading before refill
        }
#ifdef HAVE_WMMA4
        float total = c[0];                         // D row 0 = column sums
#else
        float total = accA + __shfl_xor(accA, 16, 32);
#endif
        total += fexp(alpha + ul[M_]);              // dustbin row
        int n = colBase + ncol;
        if (lane < 16)
            v[(size_t)b * (N_ + 1) + n] =
                lognu[(size_t)b * (N_ + 1) + n] - flog(total);
    } else if (wave == 0) {                          // dustbin column n = N
        float acc = 0.f;
        for (int m = lane; m <= M_; m += 32) acc += fexp(alpha + ul[m]);
        for (int o = 16; o > 0; o >>= 1) acc += __shfl_xor(acc, o, 32);
        if (lane == 0)
            v[(size_t)b * (N_ + 1) + N_] =
                lognu[(size_t)b * (N_ + 1) + N_] - flog(acc);
    }
}

// ------------------------------------------------------------ finalize ----
__global__ __launch_bounds__(256)
void ot_finalize(const float* __restrict__ scores,
                 const unsigned char* __restrict__ rowm,
                 const unsigned char* __restrict__ colm,
                 const float* __restrict__ alpha_p,
                 const float* __restrict__ u, const float* __restrict__ v,
                 const float* __restrict__ normv, float* __restrict__ out) {
    int b = blockIdx.y;
    long idx = (long)blockIdx.x * 256 + threadIdx.x;
    const long MN = (long)(M_ + 1) * (N_ + 1);
    if (idx >= MN) return;
    int i = (int)(idx / (N_ + 1));
    int j = (int)(idx % (N_ + 1));
    float z;
    bool invalid;
    if (i < M_ && j < N_) {
        z = scores[((size_t)b * M_ + i) * N_ + j];
        invalid = !rowm[b * M_ + i] || !colm[b * N_ + j];
    } else {
        z = *alpha_p;
        invalid = (i < M_ && !rowm[b * M_ + i]) || (j < N_ && !colm[b * N_ + j]);
    }
    if (invalid) z = -INF_;
    out[(size_t)b * MN + idx] = z + u[(size_t)b * (M_ + 1) + i]
                                  + v[(size_t)b * (N_ + 1) + j] - normv[b];
}

// ------------------------------------------------------------- launch -----
extern "C" void kernel_launch(void* const* d_in, const int* in_sizes, int n_in,
                              void* d_out, int out_size, void* d_ws, size_t ws_size,
                              hipStream_t stream) {
    const float* scores = (const float*)d_in[0];
    const unsigned char* rowm = (const unsigned char*)d_in[1];
    const unsigned char* colm = (const unsigned char*)d_in[2];
    const float* alpha = (const float*)d_in[3];

    float* ws = (float*)d_ws;
    float* u     = ws;                      // B*(M+1)
    float* v     = u     + B_ * (M_ + 1);   // B*(N+1)
    float* logmu = v     + B_ * (N_ + 1);
    float* lognu = logmu + B_ * (M_ + 1);
    float* normv = lognu + B_ * (N_ + 1);   // B

    ot_init<<<B_, 256, 0, stream>>>(rowm, colm, u, v, logmu, lognu, normv);
    for (int it = 0; it < 100; ++it) {
        ot_row_pass<<<dim3(M_ / 8 + 1, B_), 256, 0, stream>>>(
            scores, rowm, colm, alpha, v, logmu, u);
        ot_col_pass<<<dim3(N_ / 128 + 1, B_), 256, 0, stream>>>(
            scores, rowm, colm, alpha, u, lognu, v);
    }
    const long MN = (long)(M_ + 1) * (N_ + 1);
    ot_finalize<<<dim3((unsigned)((MN + 255) / 256), B_), 256, 0, stream>>>(
        scores, rowm, colm, alpha, u, v, normv, (float*)d_out);
}